// MVCNN_51926154609078
// MI455X (gfx1250) — compile-verified
//
#include <hip/hip_runtime.h>
#include <float.h>

typedef float v2f __attribute__((ext_vector_type(2)));
typedef float v8f __attribute__((ext_vector_type(8)));

#define BB    64
#define CC    2048
#define VV    256
#define DD    4096   // C + ART
#define FF    4096
#define OUT_N 1000
#define KS    32     // K rows staged per chunk
#define BN    32     // N columns per block

// ---------------------------------------------------------------------------
// Kernel 1: ragged max over views. One wave32 per output row (b, d).
// Pure HBM stream: 256 MB in, 1 MB out.
// ---------------------------------------------------------------------------
__global__ __launch_bounds__(256) void mvcnn_pool(const float* __restrict__ x,
                                                  const float* __restrict__ xart,
                                                  const int*   __restrict__ lens,
                                                  float*       __restrict__ pooled)
{
    const int wave = (blockIdx.x * blockDim.x + threadIdx.x) >> 5;
    const int lane = threadIdx.x & 31;
    const int b = wave >> 12;        // 4096 rows per batch element
    const int d = wave & 4095;

    const float* src = (d < CC) ? (x    + ((size_t)b * CC + d)        * VV)
                                : (xart + ((size_t)b * CC + (d - CC)) * VV);
    int len = lens[b];
    if (len < 1) len = 1;

    const float4 p0 = reinterpret_cast<const float4*>(src)[lane];       // v = 4*lane
    const float4 p1 = reinterpret_cast<const float4*>(src)[lane + 32];  // v = 128 + 4*lane
    const int v0 = lane * 4;
    const int v1 = 128 + lane * 4;

    float m = -FLT_MAX;
    m = (v0 + 0 < len) ? fmaxf(m, p0.x) : m;
    m = (v0 + 1 < len) ? fmaxf(m, p0.y) : m;
    m = (v0 + 2 < len) ? fmaxf(m, p0.z) : m;
    m = (v0 + 3 < len) ? fmaxf(m, p0.w) : m;
    m = (v1 + 0 < len) ? fmaxf(m, p1.x) : m;
    m = (v1 + 1 < len) ? fmaxf(m, p1.y) : m;
    m = (v1 + 2 < len) ? fmaxf(m, p1.z) : m;
    m = (v1 + 3 < len) ? fmaxf(m, p1.w) : m;

    #pragma unroll
    for (int off = 16; off > 0; off >>= 1)
        m = fmaxf(m, __shfl_xor(m, off, 32));

    if (lane == 0)
        pooled[wave] = m;   // wave == b*DD + d, matches [B, D] GEMM A layout
}

// ---------------------------------------------------------------------------
// Kernel 2: block-cooperative skinny GEMM  out[64, N] = act(A[64,K] @ W[K,N] + b)
//
// Block = 256 threads = 8 waves = 4 m-tiles x 2 n-subtiles -> 64x32 out strip.
// Weight chunk W[k..k+KS)[n0..n0+BN) (4 KB) is staged global->LDS with
// GLOBAL_LOAD_ASYNC_TO_LDS_B128 (16 B/thread), double-buffered so the async
// engine copies chunk c+1 while the waves run 8x v_wmma_f32_16x16x4_f32 on
// chunk c. Each weight element is read from HBM exactly once.
//
// Per ISA 7.12.2 f32 layouts (wave32, half = lane>>4, l = lane&15):
//   A 16x4 : lane holds A[m0+l][k+2*half], A[m0+l][k+2*half+1]  (float2, global)
//   B 4x16 : lane holds W[k+2*half][col],  W[k+2*half+1][col]   (ds_load from LDS)
//   C 16x16: c[r] -> out[m0 + r + 8*half][col]
// ---------------------------------------------------------------------------
template <bool RELU>
__global__ __launch_bounds__(256) void mvcnn_gemm(const float* __restrict__ A,
                                                  const float* __restrict__ W,
                                                  const float* __restrict__ bias,
                                                  float*       __restrict__ out,
                                                  int K, int N)
{
    // Single __shared__ object -> LDS offset 0; raw byte offsets used in the
    // async-copy VDST match the compiler's ds addressing of this array.
    __shared__ float smem[2][KS * BN];   // 2 x 4 KB double buffer

    const int t    = threadIdx.x;
    const int lane = t & 31;
    const int wv   = t >> 5;             // 0..7
    const int half = lane >> 4;          // 0: K{0,1}, 1: K{2,3}
    const int l    = lane & 15;
    const int mt   = wv & 3;             // m-tile 0..3
    const int ns   = wv >> 2;            // n-subtile 0..1
    const int n0   = blockIdx.x * BN;

    // async-copy mapping: thread t moves 16 B: row = t>>3, 4 cols at (t&7)*4
    const int crow = t >> 3;                       // 0..31 (= KS rows)
    int ccol = n0 + ((t & 7) << 2);
    if (ccol > N - 4) ccol = N - 4;                // clamp tail block, 16B granule
    const unsigned ldsoff = (unsigned)((crow * BN + ((t & 7) << 2)) * 4);

    const float* arow = A + (size_t)(mt * 16 + l) * K + 2 * half;
    const int ncol = ns * 16 + l;                  // column within the block strip
    const int col  = n0 + ncol;

    v8f c = {};
    const int nchunks = K / KS;

    // prologue: stage chunk 0 into buffer 0
    {
        const float* g = W + (size_t)crow * N + ccol;
        asm volatile("global_load_async_to_lds_b128 %0, %1, off"
                     :: "v"(ldsoff), "v"(g) : "memory");
    }

    for (int ch = 0; ch < nchunks; ++ch) {
        const int cur = ch & 1;
        if (ch + 1 < nchunks) {
            const float* g = W + ((size_t)(ch + 1) * KS + crow) * N + ccol;
            const unsigned dst = ldsoff + (unsigned)(((ch + 1) & 1) * (KS * BN * 4));
            asm volatile("global_load_async_to_lds_b128 %0, %1, off"
                         :: "v"(dst), "v"(g) : "memory");
            asm volatile("s_wait_asynccnt 0x1" ::: "memory");  // chunk ch landed
        } else {
            asm volatile("s_wait_asynccnt 0x0" ::: "memory");  // last chunk landed
        }
        __syncthreads();   // all waves' portions of chunk ch visible

        const float* ap = arow + ch * KS;
        #pragma unroll
        for (int kk = 0; kk < KS; kk += 4) {
            v2f a = *reinterpret_cast<const v2f*>(ap + kk);    // global_load_b64
            v2f b;
            b.x = smem[cur][(kk + 2 * half)     * BN + ncol];  // ds_load_b32
            b.y = smem[cur][(kk + 2 * half + 1) * BN + ncol];
            c = __builtin_amdgcn_wmma_f32_16x16x4_f32(
                    false, a, false, b, (short)0, c, false, false);
        }
        __syncthreads();   // done reading buf[cur] before iter ch+1 overwrites it
    }

    const int   colc = (col < N) ? col : (N - 1);
    const float bv   = bias[colc];
    #pragma unroll
    for (int r = 0; r < 8; ++r) {
        float o = c[r] + bv;
        if (RELU) o = fmaxf(o, 0.0f);
        if (col < N)
            out[(size_t)(mt * 16 + 8 * half + r) * N + col] = o;
    }
}

// ---------------------------------------------------------------------------
extern "C" void kernel_launch(void* const* d_in, const int* in_sizes, int n_in,
                              void* d_out, int out_size, void* d_ws, size_t ws_size,
                              hipStream_t stream)
{
    const float* x    = (const float*)d_in[0];
    const float* xart = (const float*)d_in[1];
    const int*   lens = (const int*)  d_in[2];
    const float* w1   = (const float*)d_in[3];
    const float* b1   = (const float*)d_in[4];
    const float* w2   = (const float*)d_in[5];
    const float* b2   = (const float*)d_in[6];
    const float* w3   = (const float*)d_in[7];
    const float* b3   = (const float*)d_in[8];

    float* ws     = (float*)d_ws;
    float* pooled = ws;                    // [64, 4096]  1 MB
    float* h1     = ws + (size_t)BB * DD;  // [64, 4096]  1 MB
    float* h2     = h1 + (size_t)BB * FF;  // [64, 4096]  1 MB

    // 1) ragged max pool: 262144 rows, 8 waves (rows) per 256-thread block
    mvcnn_pool<<<(BB * DD) / 8, 256, 0, stream>>>(x, xart, lens, pooled);

    // 2) h1 = relu(pooled @ w1 + b1)    128 blocks x 64x32 strips
    mvcnn_gemm<true><<<FF / BN, 256, 0, stream>>>(pooled, w1, b1, h1, DD, FF);

    // 3) h2 = relu(h1 @ w2 + b2)
    mvcnn_gemm<true><<<FF / BN, 256, 0, stream>>>(h1, w2, b2, h2, FF, FF);

    // 4) out = h2 @ w3 + b3   (N = 1000, tail block clamped/guarded)
    mvcnn_gemm<false><<<(OUT_N + BN - 1) / BN, 256, 0, stream>>>(h2, w3, b3,
                                                                 (float*)d_out,
                                                                 FF, OUT_N);
}